// EquiformerBlock_37873021616387
// MI455X (gfx1250) — compile-verified
//
#include <hip/hip_runtime.h>

typedef __attribute__((ext_vector_type(2))) float v2f;
typedef __attribute__((ext_vector_type(8))) float v8f;

#define DEVINL static __device__ __forceinline__

constexpr float RS3   = 0.57735026918962580f;  // 1/sqrt(3)
constexpr float RS5   = 0.44721359549995790f;  // 1/sqrt(5)
constexpr float RS32  = 0.17677669529663687f;  // 1/sqrt(32)
constexpr float RS112 = 0.09449111825230679f;  // 1/sqrt(112)
constexpr float RS128 = 0.08838834764831843f;  // 1/sqrt(128)

DEVINL float sigmoid_(float x) { return 1.0f / (1.0f + __expf(-x)); }
DEVINL float silu_(float x) { return x * sigmoid_(x); }
// monotone float <-> uint mapping for atomicMax on floats
DEVINL unsigned fmap_(float f) { unsigned u = __float_as_uint(f); return (u & 0x80000000u) ? ~u : (u | 0x80000000u); }
DEVINL float funmap_(unsigned u) { return (u & 0x80000000u) ? __uint_as_float(u ^ 0x80000000u) : __uint_as_float(~u); }

DEVINL void cpw(float* dst, const float* src, int n) {
  for (int i = (int)threadIdx.x; i < n; i += (int)blockDim.x) dst[i] = src[i];
}
// stage W (row-major K x N) transposed into LDS as WT[n*(K+2) + k]
// (+2 pad: even stride -> aligned b64 pairs; stride mod 64 in {2,18,34,50} -> no bank conflicts)
DEVINL void cpwT(float* dst, const float* src, int K, int N) {
  const int ld = K + 2;
  for (int i = (int)threadIdx.x; i < K * N; i += (int)blockDim.x) {
    const int k = i / N, n = i - k * N;
    dst[n * ld + k] = src[i];
  }
}

// One wave: Y[16 x NT*16] = act((A[16 x K] @ W) * scale + bias), cols [c0, c0+NT*16).
// A element (m,k) at A[m*lda + k*astr].
// BT=true : W transposed in LDS, element (k,n) at W[n*ldw + k]  (contiguous k pair -> one DS op)
// BT=false: W row-major (global), element (k,n) at W[k*ldw + n]
template <int ACT, int NT, bool BT, typename PA, typename PW, typename PY, typename PB>
DEVINL void gemm16_blk(PA A, int lda, int astr, PW W, int ldw,
                       PY Y, int ldy, int ystr, int K, int c0, float scale, PB bias)
{
  const int lane = (int)(threadIdx.x & 31u);
  const int half = lane >> 4;
  const int m16  = lane & 15;
  v8f acc[NT];
#pragma unroll
  for (int t = 0; t < NT; ++t) acc[t] = (v8f){0.f,0.f,0.f,0.f,0.f,0.f,0.f,0.f};
  for (int k0 = 0; k0 < K; k0 += 4) {
    const int ka = k0 + 2 * half;
    v2f a;
    a.x = A[m16 * lda + ka * astr];
    a.y = A[m16 * lda + (ka + 1) * astr];
#pragma unroll
    for (int t = 0; t < NT; ++t) {
      const int cc = c0 + t * 16 + m16;
      v2f b;
      if (BT) { b.x = W[cc * ldw + ka];     b.y = W[cc * ldw + ka + 1]; }
      else    { b.x = W[ka * ldw + cc];     b.y = W[(ka + 1) * ldw + cc]; }
      acc[t] = __builtin_amdgcn_wmma_f32_16x16x4_f32(false, a, false, b, (short)0, acc[t], false, false);
    }
  }
#pragma unroll
  for (int t = 0; t < NT; ++t) {
    const int cc = c0 + t * 16 + m16;
    const float bv = (bias != nullptr) ? bias[cc] : 0.0f;
#pragma unroll
    for (int r = 0; r < 8; ++r) {
      float v = acc[t][r] * scale + bv;
      if (ACT == 1) v = silu_(v);
      Y[(r + 8 * half) * ldy + cc * ystr] = v;
    }
  }
}

template <int ACT, bool BT, typename PA, typename PW, typename PY, typename PB>
DEVINL void gemm16(PA A, int lda, int astr, PW W, int ldw,
                   PY Y, int ldy, int ystr, int K, int ncols, float scale, PB bias)
{
  int c0 = 0;
  for (; c0 + 128 <= ncols; c0 += 128)
    gemm16_blk<ACT, 8, BT>(A, lda, astr, W, ldw, Y, ldy, ystr, K, c0, scale, bias);
  if (c0 + 64 <= ncols) {
    gemm16_blk<ACT, 4, BT>(A, lda, astr, W, ldw, Y, ldy, ystr, K, c0, scale, bias);
    c0 += 64;
  }
  if (c0 + 32 <= ncols) {
    gemm16_blk<ACT, 2, BT>(A, lda, astr, W, ldw, Y, ldy, ystr, K, c0, scale, bias);
    c0 += 32;
  }
  if (c0 < ncols)
    gemm16_blk<ACT, 1, BT>(A, lda, astr, W, ldw, Y, ldy, ystr, K, c0, scale, bias);
}

// ---------------------------------------------------------------- init
__global__ void __launch_bounds__(256)
init_ws_kernel(float* __restrict__ nacc, float* __restrict__ asum,
               unsigned* __restrict__ amax, int n240, int n4)
{
  const int i = blockIdx.x * 256 + threadIdx.x;
  if (i < n240) nacc[i] = 0.0f;
  if (i < n4) { asum[i] = 0.0f; amax[i] = 0x007FFFFFu; /* fmap_(-inf) */ }
}

// ---------------------------------------------------------------- node linear (m_src, m_dst)
struct N1Params { const float *sW0,*sW1,*sW2,*dW0,*dW1,*dW2,*db; };

__global__ void __launch_bounds__(128)
node_lin_kernel(N1Params P, const float* __restrict__ xsrc, const float* __restrict__ xdst,
                float* __restrict__ msrc, float* __restrict__ mdst, int N)
{
  extern __shared__ float smem[];
  constexpr int OS0=0, OS1=4224, OS2=5312, OD0=5600, OD1=9824, OD2=10912, ODB=11200, WTOT=11264;
  cpwT(smem+OS0, P.sW0, 64, 64); cpwT(smem+OS1, P.sW1, 32, 32); cpwT(smem+OS2, P.sW2, 16, 16);
  cpwT(smem+OD0, P.dW0, 64, 64); cpwT(smem+OD1, P.dW1, 32, 32); cpwT(smem+OD2, P.dW2, 16, 16);
  cpw(smem+ODB, P.db, 64);
  __syncthreads();
  const int wave = (int)(threadIdx.x >> 5), lane = (int)(threadIdx.x & 31u);
  const int n0 = (blockIdx.x * 4 + wave) * 16;
  if (n0 >= N) return;
  float* IN  = smem + WTOT + wave * 7744;  // 16 x 242
  float* OUT = IN + 3872;                  // 16 x 242
  for (int pass = 0; pass < 2; ++pass) {
    const float* x = pass ? xdst : xsrc;
    const int o0 = pass ? OD0 : OS0, o1 = pass ? OD1 : OS1, o2 = pass ? OD2 : OS2;
    for (int r = 0; r < 16; ++r)
      for (int c = lane; c < 240; c += 32) IN[r*242 + c] = x[(size_t)(n0 + r) * 240 + c];
    gemm16<0, true>(IN, 242, 1, smem + o0, 66, OUT, 242, 1, 64, 64, 0.125f,
                    pass ? (const float*)(smem + ODB) : (const float*)nullptr);
    for (int m = 0; m < 3; ++m)
      gemm16<0, true>(IN + 64 + m, 242, 3, smem + o1, 34, OUT + 64 + m, 242, 3, 32, 32, RS32, (const float*)nullptr);
    for (int m = 0; m < 5; ++m)
      gemm16<0, true>(IN + 160 + m, 242, 5, smem + o2, 18, OUT + 160 + m, 242, 5, 16, 16, 0.25f, (const float*)nullptr);
    float* dst = pass ? mdst : msrc;
    for (int r = 0; r < 16; ++r)
      for (int c = lane; c < 240; c += 32) dst[(size_t)(n0 + r) * 240 + c] = OUT[r*242 + c];
  }
}

// ---------------------------------------------------------------- edge pipeline
struct EParams {
  const float *fc1W,*fc1b,*fc2W,*fc2b,*fc3W,*fc3b,*alphaW,*alphaDot,
              *gW0,*gW1,*gW2,*vW0,*vW1,*vW2;
};

template <int PHASE>
__global__ void __launch_bounds__(128)
edge_kernel(EParams P,
            const float* __restrict__ msrc, const float* __restrict__ mdst,
            const int* __restrict__ esrc, const int* __restrict__ edst,
            const float* __restrict__ eattr, const float* __restrict__ escal,
            float* __restrict__ a_ws, unsigned* __restrict__ amax,
            const float* __restrict__ asum, float* __restrict__ nacc, int E)
{
  extern __shared__ float smem[];
  constexpr int OFC1W=0, OFC1B=4224, OFC2W=4288, OFC2B=8512, OFC3W=8576, OFC3B=19136;
  constexpr int OAW=19296, OADOT=22944;                                      // phase 0
  constexpr int OGW1=19296, OGW2=20384, OVW0=20672, OVW1=24896, OVW2=25984;  // phase 1
  constexpr int WTOT = (PHASE == 0) ? 22976 : 26272;
  cpwT(smem+OFC1W, P.fc1W, 64, 64);  cpw(smem+OFC1B, P.fc1b, 64);
  cpwT(smem+OFC2W, P.fc2W, 64, 64);  cpw(smem+OFC2B, P.fc2b, 64);
  cpwT(smem+OFC3W, P.fc3W, 64, 160); cpw(smem+OFC3B, P.fc3b, 160);
  if (PHASE == 0) {
    cpwT(smem+OAW, P.alphaW, 112, 32); cpw(smem+OADOT, P.alphaDot, 32);
  } else {
    cpwT(smem+OGW1, P.gW1, 32, 32); cpwT(smem+OGW2, P.gW2, 16, 16);
    cpwT(smem+OVW0, P.vW0, 64, 64); cpwT(smem+OVW1, P.vW1, 32, 32); cpwT(smem+OVW2, P.vW2, 16, 16);
  }
  __syncthreads();

  const int wave = (int)(threadIdx.x >> 5), lane = (int)(threadIdx.x & 31u);
  const int e0 = (blockIdx.x * 4 + wave) * 16;
  if (e0 >= E) return;
  float* A_ = smem + WTOT + wave * 13152;  // 4640: es/h (16x66) -> d (16x290) -> feat (16x242)
  float* B_ = A_ + 4640;                   // 4640: w (16x162) + alpha (16x34) -> g (16x290) -> val (16x242)
  float* C_ = B_ + 4640;                   // 3872: msg (16x242) -> softmax weights

  // edge_scalars tile
  for (int r = 0; r < 16; ++r)
    for (int c = lane; c < 64; c += 32) A_[r*66 + c] = escal[(size_t)(e0 + r) * 64 + c];
  // edge MLP
  gemm16<1, true>(A_,      66, 1, smem+OFC1W, 66, A_+1056, 66, 1, 64,  64, 1.0f, smem+OFC1B);
  gemm16<1, true>(A_+1056, 66, 1, smem+OFC2W, 66, A_,      66, 1, 64,  64, 1.0f, smem+OFC2B);
  gemm16<0, true>(A_,      66, 1, smem+OFC3W, 66, B_,     162, 1, 64, 160, 1.0f, smem+OFC3B);
  // message gather (node tables live in L2)
  for (int r = 0; r < 16; ++r) {
    const int e = e0 + r;
    const size_t sb = (size_t)esrc[e] * 240, db = (size_t)edst[e] * 240;
    for (int c = lane; c < 240; c += 32) C_[r*242 + c] = msrc[sb + c] + mdst[db + c];
  }
  // dtp: (msg, edge_attr, w) -> d (16 x 288, ld 290)
  for (int r = 0; r < 16; ++r) {
    const int e = e0 + r;
    const float* mg = C_ + r*242;
    const float* wr = B_ + r*162;
    float* dr = A_ + r*290;
    const float ea0 = eattr[(size_t)e*9+0];
    const float ex = eattr[(size_t)e*9+1], ey = eattr[(size_t)e*9+2], ez = eattr[(size_t)e*9+3];
    const float q0 = eattr[(size_t)e*9+4], q1 = eattr[(size_t)e*9+5], q2 = eattr[(size_t)e*9+6],
                q3 = eattr[(size_t)e*9+7], q4 = eattr[(size_t)e*9+8];
    for (int c = lane; c < 288; c += 32) {
      float v;
      if (c < 64) {
        v = mg[c] * ea0 * wr[c];
      } else if (c < 96) {
        const int i = c - 64;
        v = (mg[64+i*3]*ex + mg[64+i*3+1]*ey + mg[64+i*3+2]*ez) * wr[112+i] * RS3;
      } else if (c < 112) {
        const int i = c - 96;
        v = (mg[160+i*5]*q0 + mg[160+i*5+1]*q1 + mg[160+i*5+2]*q2
           + mg[160+i*5+3]*q3 + mg[160+i*5+4]*q4) * wr[144+i] * RS5;
      } else if (c < 208) {
        const int j = c - 112;
        v = mg[64+j] * ea0 * wr[64 + j/3];
      } else {
        const int j = c - 208;
        v = mg[160+j] * ea0 * wr[96 + j/5];
      }
      dr[c] = v;
    }
  }

  if (PHASE == 0) {
    // alpha logits: (d[:, :112] @ alpha_W)/sqrt(112) -> leaky_relu -> dot alpha_dot
    gemm16<0, true>(A_, 290, 1, smem+OAW, 114, B_+2592, 34, 1, 112, 32, RS112, (const float*)nullptr);
    if (lane < 16) {
      const int e = e0 + lane;
      const float* ar = B_ + 2592 + lane*34;
      const int nb = edst[e] * 4;
      for (int h = 0; h < 4; ++h) {
        float s = 0.f;
        for (int j = 0; j < 8; ++j) {
          float x = ar[h*8 + j];
          x = (x > 0.f) ? x : 0.2f * x;
          s += x * smem[OADOT + h*8 + j];
        }
        a_ws[(size_t)e*4 + h] = s;
        atomicMax(&amax[nb + h], fmap_(s));
      }
    }
  } else {
    // g = linear(d, DTP_OUT -> GATED_ATT); g_W0 streamed from global (L2-resident)
    gemm16<0, false>(A_, 290, 1, P.gW0, 112, B_, 290, 1, 112, 112, RS112, (const float*)nullptr);
    for (int m = 0; m < 3; ++m)
      gemm16<0, true>(A_+112+m, 290, 3, smem+OGW1, 34, B_+112+m, 290, 3, 32, 32, RS32, (const float*)nullptr);
    for (int m = 0; m < 5; ++m)
      gemm16<0, true>(A_+208+m, 290, 5, smem+OGW2, 18, B_+208+m, 290, 5, 16, 16, 0.25f, (const float*)nullptr);
    // feat = gate(g, 64, 48) -> A_ (16 x 242), d is dead
    for (int r = 0; r < 16; ++r) {
      const float* gr = B_ + r*290;
      float* fr = A_ + r*242;
      for (int c = lane; c < 240; c += 32) {
        float v;
        if (c < 64)        v = silu_(gr[c]);
        else if (c < 160)  { const int j = c-64;  v = gr[112+j] * sigmoid_(gr[64 + j/3]); }
        else               { const int j = c-160; v = gr[208+j] * sigmoid_(gr[96 + j/5]); }
        fr[c] = v;
      }
    }
    // val = linear(feat, EMB->EMB) -> B_ (g is dead)
    gemm16<0, true>(A_, 242, 1, smem+OVW0, 66, B_, 242, 1, 64, 64, 0.125f, (const float*)nullptr);
    for (int m = 0; m < 3; ++m)
      gemm16<0, true>(A_+64+m, 242, 3, smem+OVW1, 34, B_+64+m, 242, 3, 32, 32, RS32, (const float*)nullptr);
    for (int m = 0; m < 5; ++m)
      gemm16<0, true>(A_+160+m, 242, 5, smem+OVW2, 18, B_+160+m, 242, 5, 16, 16, 0.25f, (const float*)nullptr);
    // per-(row, head) softmax weights -> C_[0:64] (msg dead)
    for (int t = lane; t < 64; t += 32) {
      const int r = t >> 2, h = t & 3;
      const int e = e0 + r;
      const int nb = edst[e]*4 + h;
      C_[t] = __expf(a_ws[(size_t)e*4 + h] - funmap_(amax[nb])) / (asum[nb] + 1e-9f);
    }
    // weighted scatter-add into node accumulators
    for (int r = 0; r < 16; ++r) {
      const int e = e0 + r;
      float* nb = nacc + (size_t)edst[e] * 240;
      const float* vr = B_ + r*242;
      for (int c = lane; c < 240; c += 32) {
        int h;
        if (c < 64)       h = c >> 4;
        else if (c < 160) h = ((c - 64) / 3) >> 3;
        else              h = ((c - 160) / 5) >> 2;
        atomicAdd(&nb[c], vr[c] * C_[r*4 + h]);
      }
    }
  }
}

// ---------------------------------------------------------------- softmax denominator
__global__ void __launch_bounds__(256)
softmax_sum_kernel(const float* __restrict__ a_ws, const unsigned* __restrict__ amax,
                   float* __restrict__ asum, const int* __restrict__ edst, int E4)
{
  const int i = blockIdx.x * 256 + threadIdx.x;
  if (i >= E4) return;
  const int nb = edst[i >> 2] * 4 + (i & 3);
  atomicAdd(&asum[nb], __expf(a_ws[i] - funmap_(amax[nb])));
}

// ---------------------------------------------------------------- node tail (proj/ln/FFN)
struct N2Params {
  const float *pW0,*pW1,*pW2,*pb,*lnw,*lnb,*f1W0,*f1W1,*f1W2,*f1b,*f2W0,*f2W1,*f2W2,*f2b;
};

__global__ void __launch_bounds__(64)
node_out_kernel(N2Params P, const float* __restrict__ xdst, const float* __restrict__ nacc,
                float* __restrict__ out, int N)
{
  extern __shared__ float smem[];
  constexpr int OP0=0, OP1=4224, OP2=5312, OPB=5600, OLW=5664, OLB=5776,
                O11=5840, O12=8016, O1B=8592, O20=8816, O21=17136, O22=19248, O2B=19792, WTOT=19856;
  cpwT(smem+OP0, P.pW0, 64, 64); cpwT(smem+OP1, P.pW1, 32, 32); cpwT(smem+OP2, P.pW2, 16, 16);
  cpw(smem+OPB, P.pb, 64);  cpw(smem+OLW, P.lnw, 112); cpw(smem+OLB, P.lnb, 64);
  cpwT(smem+O11, P.f1W1, 32, 64); cpwT(smem+O12, P.f1W2, 16, 32); cpw(smem+O1B, P.f1b, 224);
  cpwT(smem+O20, P.f2W0, 128, 64); cpwT(smem+O21, P.f2W1, 64, 32); cpwT(smem+O22, P.f2W2, 32, 16);
  cpw(smem+O2B, P.f2b, 64);
  __syncthreads();
  const int wave = (int)(threadIdx.x >> 5), lane = (int)(threadIdx.x & 31u);
  const int n0 = (blockIdx.x * 2 + wave) * 16;
  if (n0 >= N) return;
  float* A_ = smem + WTOT + wave * 28576;  // nacc tile -> final y2 (16x242)
  float* B_ = A_ + 3872;                   // node_output (16x242)
  float* Cc = B_ + 3872;                   // proj tmp -> LN out (16x242)
  float* D_ = Cc + 3872;                   // f1 out (16x578)
  float* G_ = D_ + 9248;                   // gated (16x482)

  for (int r = 0; r < 16; ++r)
    for (int c = lane; c < 240; c += 32) A_[r*242 + c] = nacc[(size_t)(n0 + r) * 240 + c];
  // proj (EMB->EMB, bias on l0)
  gemm16<0, true>(A_, 242, 1, smem+OP0, 66, Cc, 242, 1, 64, 64, 0.125f, smem+OPB);
  for (int m = 0; m < 3; ++m)
    gemm16<0, true>(A_+64+m, 242, 3, smem+OP1, 34, Cc+64+m, 242, 3, 32, 32, RS32, (const float*)nullptr);
  for (int m = 0; m < 5; ++m)
    gemm16<0, true>(A_+160+m, 242, 5, smem+OP2, 18, Cc+160+m, 242, 5, 16, 16, 0.25f, (const float*)nullptr);
  // residual
  for (int r = 0; r < 16; ++r)
    for (int c = lane; c < 240; c += 32)
      B_[r*242 + c] = xdst[(size_t)(n0 + r) * 240 + c] + Cc[r*242 + c];
  // equivariant layer norm -> Cc
  if (lane < 16) {
    const float* f = B_ + lane*242;
    float* y = Cc + lane*242;
    float mu = 0.f;
    for (int i = 0; i < 64; ++i) mu += f[i];
    mu *= (1.f/64.f);
    float n2 = 0.f;
    for (int i = 0; i < 64; ++i) { const float t = f[i] - mu; n2 += t*t; }
    float rq = rsqrtf(n2 * (1.f/64.f) + 1e-5f);
    for (int i = 0; i < 64; ++i) y[i] = (f[i]-mu)*rq*smem[OLW+i] + smem[OLB+i];
    n2 = 0.f;
    for (int i = 0; i < 96; ++i) { const float t = f[64+i]; n2 += t*t; }
    rq = rsqrtf(n2 * (1.f/32.f) + 1e-5f);
    for (int i = 0; i < 32; ++i)
      for (int m = 0; m < 3; ++m) y[64+i*3+m] = f[64+i*3+m]*rq*smem[OLW+64+i];
    n2 = 0.f;
    for (int i = 0; i < 80; ++i) { const float t = f[160+i]; n2 += t*t; }
    rq = rsqrtf(n2 * (1.f/16.f) + 1e-5f);
    for (int i = 0; i < 16; ++i)
      for (int m = 0; m < 5; ++m) y[160+i*5+m] = f[160+i*5+m]*rq*smem[OLW+96+i];
  }
  // f1 (EMB -> MID_GATED); f1_W0 streamed from global
  gemm16<0, false>(Cc, 242, 1, P.f1W0, 224, D_, 578, 1, 64, 224, 0.125f, smem+O1B);
  for (int m = 0; m < 3; ++m)
    gemm16<0, true>(Cc+64+m, 242, 3, smem+O11, 34, D_+224+m, 578, 3, 32, 64, RS32, (const float*)nullptr);
  for (int m = 0; m < 5; ++m)
    gemm16<0, true>(Cc+160+m, 242, 5, smem+O12, 18, D_+416+m, 578, 5, 16, 32, 0.25f, (const float*)nullptr);
  // gate(y, 128, 96) -> G_
  for (int r = 0; r < 16; ++r) {
    const float* yr = D_ + r*578;
    float* gr = G_ + r*482;
    for (int c = lane; c < 480; c += 32) {
      float v;
      if (c < 128)      v = silu_(yr[c]);
      else if (c < 320) { const int j = c-128; v = yr[224+j] * sigmoid_(yr[128 + j/3]); }
      else              { const int j = c-320; v = yr[416+j] * sigmoid_(yr[192 + j/5]); }
      gr[c] = v;
    }
  }
  // f2 (MID -> EMB) -> A_
  gemm16<0, true>(G_, 482, 1, smem+O20, 130, A_, 242, 1, 128, 64, RS128, smem+O2B);
  for (int m = 0; m < 3; ++m)
    gemm16<0, true>(G_+128+m, 482, 3, smem+O21, 66, A_+64+m, 242, 3, 64, 32, 0.125f, (const float*)nullptr);
  for (int m = 0; m < 5; ++m)
    gemm16<0, true>(G_+320+m, 482, 5, smem+O22, 34, A_+160+m, 242, 5, 32, 16, RS32, (const float*)nullptr);
  // final residual
  for (int r = 0; r < 16; ++r)
    for (int c = lane; c < 240; c += 32)
      out[(size_t)(n0 + r) * 240 + c] = B_[r*242 + c] + A_[r*242 + c];
}

// ---------------------------------------------------------------- host
extern "C" void kernel_launch(void* const* d_in, const int* in_sizes, int n_in,
                              void* d_out, int out_size, void* d_ws, size_t ws_size,
                              hipStream_t stream)
{
  const float* xsrc  = (const float*)d_in[0];
  const float* xdst  = (const float*)d_in[1];
  const int*   esrc  = (const int*)d_in[3];
  const int*   edst  = (const int*)d_in[4];
  const float* eattr = (const float*)d_in[5];
  const float* escal = (const float*)d_in[6];
  const int pbase = n_in - 35;  // 35 param tensors, flattened in dict order
  const float* prm[35];
  for (int i = 0; i < 35; ++i) prm[i] = (const float*)d_in[pbase + i];

  const int N = in_sizes[0] / 240;
  const int E = in_sizes[3];

  float* ws = (float*)d_ws;
  float*    msrc = ws;
  float*    mdst = msrc + (size_t)N * 240;
  float*    a_ws = mdst + (size_t)N * 240;
  unsigned* amax = (unsigned*)(a_ws + (size_t)E * 4);
  float*    asum = (float*)(amax + (size_t)N * 4);
  float*    nacc = asum + (size_t)N * 4;

  const int n240 = N * 240, n4 = N * 4;
  init_ws_kernel<<<(n240 + 255) / 256, 256, 0, stream>>>(nacc, asum, amax, n240, n4);

  N1Params p1{prm[0], prm[1], prm[2], prm[3], prm[4], prm[5], prm[6]};
  const size_t lds1 = (size_t)(11264 + 4 * 7744) * sizeof(float);
  node_lin_kernel<<<(N + 63) / 64, 128, lds1, stream>>>(p1, xsrc, xdst, msrc, mdst, N);

  EParams pe{prm[7],  prm[8],  prm[9],  prm[10], prm[11], prm[12], prm[13],
             prm[14], prm[15], prm[16], prm[17], prm[18], prm[19], prm[20]};
  const size_t lds_e0 = (size_t)(22976 + 4 * 13152) * sizeof(float);
  const size_t lds_e1 = (size_t)(26272 + 4 * 13152) * sizeof(float);
  edge_kernel<0><<<(E + 63) / 64, 128, lds_e0, stream>>>(
      pe, msrc, mdst, esrc, edst, eattr, escal, a_ws, amax, asum, nacc, E);
  softmax_sum_kernel<<<(E * 4 + 255) / 256, 256, 0, stream>>>(a_ws, amax, asum, edst, E * 4);
  edge_kernel<1><<<(E + 63) / 64, 128, lds_e1, stream>>>(
      pe, msrc, mdst, esrc, edst, eattr, escal, a_ws, amax, asum, nacc, E);

  N2Params p2{prm[21], prm[22], prm[23], prm[24], prm[25], prm[26], prm[27],
              prm[28], prm[29], prm[30], prm[31], prm[32], prm[33], prm[34]};
  const size_t lds2 = (size_t)(19856 + 2 * 28576) * sizeof(float);
  node_out_kernel<<<(N + 31) / 32, 64, lds2, stream>>>(p2, xdst, nacc, (float*)d_out, N);
}